// LocalAttention_57380763074827
// MI455X (gfx1250) — compile-verified
//
#include <hip/hip_runtime.h>

// CDNA5 / gfx1250 fp32 WMMA types
typedef float v2f __attribute__((ext_vector_type(2)));
typedef float v8f __attribute__((ext_vector_type(8)));

#define B_    8
#define S_    4096
#define D_    64
#define HALF_ 32
#define W_    65
#define MIN_SCORE_ (-1000000000.0f)

#define QT     16                 // queries per wave (M tile)
#define WAVES  2                  // waves per block
#define QBLK   (QT * WAVES)       // 32 queries per block
#define KROWS  (QBLK + 2 * HALF_) // 96 staged K/V rows per block
#define KSTR   68                 // LDS row stride (floats) for K/V
#define PSTR   82                 // LDS row stride (floats) for score/prob tile
#define PCOLS  80                 // 5 column tiles of 16 per wave
#define NCT    5

__global__ __launch_bounds__(QBLK * 2) void local_attn_wmma_f32(
    const float* __restrict__ Q, const float* __restrict__ K,
    const float* __restrict__ V, const unsigned char* __restrict__ mask,
    float* __restrict__ outV, float* __restrict__ outP)
{
    __shared__ float Kbuf[KROWS * KSTR];
    __shared__ float Vbuf[KROWS * KSTR];
    __shared__ float Pbuf[WAVES * QT * PSTR];

    const int blk   = blockIdx.x;
    const int b     = blk >> 7;            // S_/QBLK = 128 groups per batch
    const int grp   = blk & 127;
    const int gbase = grp * QBLK;          // first query of this block
    const int jbase = gbase - HALF_;       // first staged key row (may be <0)

    const int tid  = threadIdx.x;
    const int wave = tid >> 5;
    const int lane = tid & 31;
    const int lo   = lane & 15;
    const int hi   = lane >> 4;

    // ---- stage K/V tile (96 rows x 64 cols) to LDS, zero-fill OOB rows ----
    for (int it = tid; it < KROWS * (D_ / 4); it += QBLK * 2) {
        int r = it >> 4;
        int c = (it & 15) * 4;
        int j = jbase + r;
        float4 kv = make_float4(0.f, 0.f, 0.f, 0.f);
        float4 vv = make_float4(0.f, 0.f, 0.f, 0.f);
        if (j >= 0 && j < S_) {
            kv = *(const float4*)(K + (size_t)(b * S_ + j) * D_ + c);
            vv = *(const float4*)(V + (size_t)(b * S_ + j) * D_ + c);
        }
        *(float4*)(&Kbuf[r * KSTR + c]) = kv;
        *(float4*)(&Vbuf[r * KSTR + c]) = vv;
    }
    __syncthreads();

    const int s0    = gbase + wave * QT;   // this wave's first query
    const int krow0 = wave * QT;           // this wave's first LDS K/V row

    // ---- Q fragments: ISA 16x4 f32 A-layout.
    // lanes 0-15: M=lo, K = 4k+{0,1}; lanes 16-31: M=lo, K = 4k+{2,3}
    v2f qa[16];
    {
        const float* qrow = Q + (size_t)(b * S_ + s0 + lo) * D_ + 2 * hi;
        #pragma unroll
        for (int kk = 0; kk < 16; ++kk)
            qa[kk] = *(const v2f*)(qrow + kk * 4);
    }

    // ---- scores: S = Q (16x64) x K^T (64x80) via 5 tiles x 16 k-steps ----
    v8f sacc[NCT];
    #pragma unroll
    for (int t = 0; t < NCT; ++t) {
        #pragma unroll
        for (int r = 0; r < 8; ++r) sacc[t][r] = 0.0f;
    }

    #pragma unroll
    for (int t = 0; t < NCT; ++t) {
        #pragma unroll
        for (int kk = 0; kk < 16; ++kk) {
            // B-layout 4x16: lanes 0-15 hold K={0,1}, lanes 16-31 K={2,3}, N=lo
            v2f bk = *(const v2f*)(&Kbuf[(krow0 + t * 16 + lo) * KSTR + kk * 4 + 2 * hi]);
            sacc[t] = __builtin_amdgcn_wmma_f32_16x16x4_f32(
                false, qa[kk], false, bk, (short)0, sacc[t], false, false);
        }
    }

    // ---- spill scores to per-wave LDS tile (C-layout: VGPR r -> M = r+8*hi, N = lo) ----
    float* Pw = &Pbuf[wave * QT * PSTR];
    #pragma unroll
    for (int t = 0; t < NCT; ++t) {
        #pragma unroll
        for (int r = 0; r < 8; ++r)
            Pw[(r + 8 * hi) * PSTR + t * 16 + lo] = sacc[t][r];
    }
    __syncthreads();

    // ---- masked softmax: lane L (0..15) owns query row s0+L, 80 columns ----
    if (lane < 16) {
        const int L = lane;
        const int s = s0 + L;
        float* row = &Pw[L * PSTR];
        const unsigned char* mrow = mask + (size_t)b * S_;

        float mx = MIN_SCORE_;
        for (int c = 0; c < PCOLS; ++c) {
            int wofs = c - L;              // window index 0..64 when valid
            int j    = s0 - HALF_ + c;     // global key index
            bool ok = (wofs >= 0) & (wofs < W_) & (j >= 0) & (j < S_);
            if (ok) ok = (mrow[j] == 0);
            float sc = ok ? row[c] : MIN_SCORE_;
            mx = fmaxf(mx, sc);
        }
        float sum = 0.0f;
        for (int c = 0; c < PCOLS; ++c) {
            int wofs = c - L;
            int j    = s0 - HALF_ + c;
            bool ok = (wofs >= 0) & (wofs < W_) & (j >= 0) & (j < S_);
            if (ok) ok = (mrow[j] == 0);
            float e = ok ? __expf(row[c] - mx) : 0.0f;
            row[c] = e;
            sum += e;
        }
        const float inv = 1.0f / sum;
        float* pout = outP + (size_t)(b * S_ + s) * W_;
        for (int c = 0; c < PCOLS; ++c) {
            float p = row[c] * inv;
            row[c] = p;                    // normalized prob back to LDS for P*V
            int wofs = c - L;
            if (wofs >= 0 && wofs < W_) pout[wofs] = p;
        }
    }
    __syncthreads();

    // ---- output: O = P (16x80) x V (80x64), 20 k-steps x 4 n-tiles ----
    v8f oacc[4];
    #pragma unroll
    for (int nt = 0; nt < 4; ++nt) {
        #pragma unroll
        for (int r = 0; r < 8; ++r) oacc[nt][r] = 0.0f;
    }

    #pragma unroll
    for (int k0 = 0; k0 < PCOLS; k0 += 4) {
        // A fragment from P: lane (hi,lo): M=lo, K = k0 + 2*hi + {0,1}
        v2f ap = *(const v2f*)(&Pw[lo * PSTR + k0 + 2 * hi]);
        #pragma unroll
        for (int nt = 0; nt < 4; ++nt) {
            v2f bv;
            bv.x = Vbuf[(krow0 + k0 + 2 * hi    ) * KSTR + nt * 16 + lo];
            bv.y = Vbuf[(krow0 + k0 + 2 * hi + 1) * KSTR + nt * 16 + lo];
            oacc[nt] = __builtin_amdgcn_wmma_f32_16x16x4_f32(
                false, ap, false, bv, (short)0, oacc[nt], false, false);
        }
    }

    // ---- store D-matrix (C/D layout) straight to global ----
    #pragma unroll
    for (int nt = 0; nt < 4; ++nt) {
        #pragma unroll
        for (int r = 0; r < 8; ++r) {
            int m = r + 8 * hi;
            outV[(size_t)(b * S_ + s0 + m) * D_ + nt * 16 + lo] = oacc[nt][r];
        }
    }
}

extern "C" void kernel_launch(void* const* d_in, const int* in_sizes, int n_in,
                              void* d_out, int out_size, void* d_ws, size_t ws_size,
                              hipStream_t stream) {
    const float* Q = (const float*)d_in[0];
    const float* K = (const float*)d_in[1];
    const float* V = (const float*)d_in[2];
    const unsigned char* mask = (const unsigned char*)d_in[3];

    float* outV = (float*)d_out;                       // (B,S,D) new_value
    float* outP = (float*)d_out + (size_t)B_ * S_ * D_; // (B,S,W) probs

    dim3 grid(B_ * (S_ / QBLK));   // 8 * 128 = 1024 blocks
    dim3 block(QBLK * 2);          // 64 threads = 2 waves
    local_attn_wmma_f32<<<grid, block, 0, stream>>>(Q, K, V, mask, outV, outP);
}